// SparseEdgeConv_43903155699866
// MI455X (gfx1250) — compile-verified
//
#include <hip/hip_runtime.h>
#include <cstddef>

typedef __attribute__((ext_vector_type(2))) float v2f;
typedef __attribute__((ext_vector_type(8))) float v8f;

#define D_IO 64
#define EDIM 16
#define LDS_PITCH 66

// ---------------- zero out + cnt ----------------
__global__ void zero_kernel(float* __restrict__ out, float* __restrict__ cnt,
                            int n_out, int n_cnt) {
    int i = blockIdx.x * blockDim.x + threadIdx.x;
    if (i < n_out) out[i] = 0.0f;
    if (i < n_cnt) cnt[i] = 0.0f;
}

// ---------------- per-edge gate: sigmoid(ef . W_edge + b) ----------------
__global__ void gate_kernel(const float* __restrict__ ef,
                            const float* __restrict__ We,
                            const float* __restrict__ be,
                            float* __restrict__ gate, int E) {
    int e = blockIdx.x * blockDim.x + threadIdx.x;
    if (e >= E) return;
    const float4* p = reinterpret_cast<const float4*>(ef + (size_t)e * EDIM);
    const float4* w = reinterpret_cast<const float4*>(We);
    float s = be[0];
#pragma unroll
    for (int i = 0; i < 4; ++i) {
        float4 a = p[i];
        float4 b = w[i];
        s += a.x * b.x + a.y * b.y + a.z * b.z + a.w * b.w;
    }
    gate[e] = 1.0f / (1.0f + __expf(-s));
}

// ---------------- node projection xp = x @ W_node^T + b via WMMA f32 16x16x4 ----------------
// One wave computes a 16(node-rows) x 64(out-channels) strip.
// 4 waves / block; W_node staged in LDS with pitch 66 to avoid bank conflicts.
__global__ void __launch_bounds__(128) proj_kernel(const float* __restrict__ x,
                                                   const float* __restrict__ Wn,
                                                   const float* __restrict__ bn,
                                                   float* __restrict__ xp, int N) {
    __shared__ float w_lds[D_IO * LDS_PITCH];

    // cooperative load of W_node[64][64] into LDS (pitch 66)
    for (int i = threadIdx.x; i < D_IO * (D_IO / 4); i += 128) {
        int r  = i >> 4;           // row 0..63  (out-channel)
        int c4 = (i & 15) * 4;     // col group  (in-channel)
        float4 v = reinterpret_cast<const float4*>(Wn + (size_t)r * D_IO + c4)[0];
        float* dst = &w_lds[r * LDS_PITCH + c4];
        dst[0] = v.x; dst[1] = v.y; dst[2] = v.z; dst[3] = v.w;
    }
    __syncthreads();

    const int wave = threadIdx.x >> 5;
    const int lane = threadIdx.x & 31;
    const int tile = blockIdx.x * 4 + wave;   // tile of 16 node rows
    const int row0 = tile * 16;
    if (row0 >= N) return;

    int mrow = row0 + (lane & 15);
    if (mrow >= N) mrow = N - 1;              // N is a multiple of 16 here; safety clamp
    const int khalf = (lane >> 4) * 2;        // lanes 0-15 -> K{0,1}; lanes 16-31 -> K{2,3}
    const int ncol_lane = lane & 15;

    v8f acc[4];
#pragma unroll
    for (int t = 0; t < 4; ++t) acc[t] = (v8f){0, 0, 0, 0, 0, 0, 0, 0};

#pragma unroll
    for (int k0 = 0; k0 < D_IO; k0 += 4) {
        // A fragment (16x4 f32): lane holds x[mrow][k0+khalf], x[mrow][k0+khalf+1]
        const float* ap = x + (size_t)mrow * D_IO + k0 + khalf;
        v2f a;
        a.x = ap[0];
        a.y = ap[1];
#pragma unroll
        for (int t = 0; t < 4; ++t) {
            // B fragment (4x16 f32): B[k][n] = W_node[n][k]
            const float* bp = &w_lds[(t * 16 + ncol_lane) * LDS_PITCH + k0 + khalf];
            v2f b;
            b.x = bp[0];
            b.y = bp[1];
            acc[t] = __builtin_amdgcn_wmma_f32_16x16x4_f32(
                false, a, false, b, (short)0, acc[t], false, false);
        }
    }

    // C/D layout: VGPR v -> M = v (lanes 0-15) / v+8 (lanes 16-31), N = lane&15
#pragma unroll
    for (int t = 0; t < 4; ++t) {
        const int c = t * 16 + ncol_lane;
        const float bias = bn[c];
#pragma unroll
        for (int v = 0; v < 8; ++v) {
            const int r = row0 + v + (lane >> 4) * 8;
            if (r < N) xp[(size_t)r * D_IO + c] = acc[t][v] + bias;
        }
    }
}

// ---------------- wave-per-edge gather + atomic scatter-add ----------------
__global__ void scatter_kernel(const long long* __restrict__ ei,
                               const float* __restrict__ gate,
                               const float* __restrict__ xp,
                               float* __restrict__ out,
                               float* __restrict__ cnt, int E) {
    int gid  = blockIdx.x * blockDim.x + threadIdx.x;
    int e    = gid >> 5;
    int lane = gid & 31;
    if (e >= E) return;

    long long row = ei[e];                 // destination
    long long col = ei[(size_t)E + e];     // source
    float g = gate[e];

    const float2 v = *reinterpret_cast<const float2*>(xp + (size_t)col * D_IO + lane * 2);
    float* dst = out + (size_t)row * D_IO + lane * 2;
    atomicAdd(dst,     g * v.x);
    atomicAdd(dst + 1, g * v.y);
    if (lane == 0) atomicAdd(cnt + row, 1.0f);
}

// ---------------- mean: divide by max(cnt,1) ----------------
__global__ void finalize_kernel(float* __restrict__ out,
                                const float* __restrict__ cnt, int n_out) {
    int i = blockIdx.x * blockDim.x + threadIdx.x;
    if (i >= n_out) return;
    float c = cnt[i >> 6];
    out[i] = out[i] / fmaxf(c, 1.0f);
}

extern "C" void kernel_launch(void* const* d_in, const int* in_sizes, int n_in,
                              void* d_out, int out_size, void* d_ws, size_t ws_size,
                              hipStream_t stream) {
    const float*     x  = (const float*)d_in[0];
    const long long* ei = (const long long*)d_in[1];   // int64 per reference
    const float*     ef = (const float*)d_in[2];
    const float*     We = (const float*)d_in[3];
    const float*     be = (const float*)d_in[4];
    const float*     Wn = (const float*)d_in[5];
    const float*     bn = (const float*)d_in[6];
    float* out = (float*)d_out;

    const int N = in_sizes[0] / D_IO;      // 100000
    const int E = in_sizes[2] / EDIM;      // 1600000
    const int n_out = N * D_IO;

    // workspace layout: gate[E] | xp[N*64] | cnt[N]
    float* gate = (float*)d_ws;
    float* xp   = gate + (((size_t)E + 255) & ~(size_t)255);
    float* cnt  = xp + (size_t)N * D_IO;

    const int T = 256;
    zero_kernel<<<(n_out + T - 1) / T, T, 0, stream>>>(out, cnt, n_out, N);
    gate_kernel<<<(E + T - 1) / T, T, 0, stream>>>(ef, We, be, gate, E);

    const int ntiles = (N + 15) / 16;
    proj_kernel<<<(ntiles + 3) / 4, 128, 0, stream>>>(x, Wn, bn, xp, N);

    const long long sc_threads = (long long)E * 32;
    scatter_kernel<<<(int)((sc_threads + T - 1) / T), T, 0, stream>>>(ei, gate, xp, out, cnt, E);

    finalize_kernel<<<(n_out + T - 1) / T, T, 0, stream>>>(out, cnt, n_out);
}